// VQVAELoss_5729486373475
// MI455X (gfx1250) — compile-verified
//
#include <hip/hip_runtime.h>
#include <hip/hip_bf16.h>

// ---------------------------------------------------------------------------
// VQ-VAE loss for MI455X (gfx1250, wave32):
//   kernel 1: one workgroup per batch (64 WGs x 512 thr = 16 waves).
//             - loads x / x_hat rows into LDS, computes Huber partial
//             - tiled-wavefront soft-DTW: 16x16 tiles, 63 tile-diagonals,
//               one wave per tile; Gram tile via 4x V_WMMA_F32_16X16X4_F32
//               (exact fp32), intra-tile 31-step DP is wave-local via LDS
//               (same-wave DS ops are in-order on CDNA5 -> no barriers).
//               Steady-state DP step: 2 ds_loads + 1 ds_store (a/lf carried
//               in registers across diagonals).
//   kernel 2: reduces 64 huber + 64 sdtw partials, emits [total,huber,sdtw,c].
// Inputs are 4MB total -> fully L2 resident; problem is DP-latency bound,
// so the design minimizes workgroup barriers (63 instead of 1023) and keeps
// the serial recurrence inside a single wave.
// ---------------------------------------------------------------------------

typedef __attribute__((ext_vector_type(2))) float v2f;
typedef __attribute__((ext_vector_type(8))) float v8f;

#define LDS_FENCE() asm volatile("" ::: "memory")

constexpr int   kB = 64, kL = 512, kC = 16;
constexpr int   kNT = 32;               // 512/16 tiles per side
constexpr int   kWaves = 16, kThreads = 512;
constexpr int   kXStride = 20;          // padded row stride (floats): 16B-aligned, conflict-free
constexpr float kInf = 1.0e10f;
constexpr float kGamma = 0.1f, kInvGamma = 10.0f, kAlpha = 0.1f;

// LDS layout (floats)
constexpr int OFF_XS = 0;                          // x rows   [512][20]
constexpr int OFF_YS = OFF_XS + kL * kXStride;     // xh rows  [512][20]
constexpr int OFF_X2 = OFF_YS + kL * kXStride;     // |x_i|^2  [512]
constexpr int OFF_Y2 = OFF_X2 + kL;                // |y_j|^2  [512]
constexpr int OFF_HT = OFF_Y2 + kL;                // Htop  [32][17]  (R row boundary per tile-col)
constexpr int OFF_VL = OFF_HT + kNT * 17;          // Vleft [32][17]  (R col boundary per tile-row)
constexpr int OFF_DT = OFF_VL + kNT * 17;          // per-wave D tile [16][18]
constexpr int OFF_RT = OFF_DT + kWaves * 16 * 18;  // per-wave R tile [17][18]
constexpr int OFF_HR = OFF_RT + kWaves * 17 * 18;  // huber wave partials [16]
constexpr int SMEM_FLOATS = OFF_HR + kWaves;
constexpr size_t SMEM_BYTES = (size_t)SMEM_FLOATS * sizeof(float);

__device__ __forceinline__ float huber_term(float xx, float yy) {
    float d = yy - xx;                 // x_hat - x
    float ad = fabsf(d);
    return (ad <= 1.0f) ? 0.5f * d * d : (ad - 0.5f);
}

__launch_bounds__(kThreads)
__global__ void sdtw_huber_kernel(const float* __restrict__ x,
                                  const float* __restrict__ xh,
                                  float* __restrict__ ws)
{
    extern __shared__ float smem[];
    float* Xs    = smem + OFF_XS;
    float* Ys    = smem + OFF_YS;
    float* x2s   = smem + OFF_X2;
    float* y2s   = smem + OFF_Y2;
    float* Htop  = smem + OFF_HT;
    float* Vleft = smem + OFF_VL;
    float* Dts   = smem + OFF_DT;
    float* Rt    = smem + OFF_RT;
    float* hred  = smem + OFF_HR;

    const int b    = blockIdx.x;
    const int tid  = threadIdx.x;
    const int wave = tid >> 5;
    const int lane = tid & 31;

    // ---- stage rows into LDS; fused x2/y2 + Huber partial -----------------
    {
        const float* xrow = x  + ((size_t)b * kL + tid) * kC;
        const float* yrow = xh + ((size_t)b * kL + tid) * kC;
        float hub = 0.f, sx2 = 0.f, sy2 = 0.f;
#pragma unroll
        for (int k = 0; k < kC; k += 4) {
            float4 xv = *(const float4*)(xrow + k);
            float4 yv = *(const float4*)(yrow + k);
            *(float4*)(Xs + tid * kXStride + k) = xv;
            *(float4*)(Ys + tid * kXStride + k) = yv;
            sx2 += xv.x*xv.x + xv.y*xv.y + xv.z*xv.z + xv.w*xv.w;
            sy2 += yv.x*yv.x + yv.y*yv.y + yv.z*yv.z + yv.w*yv.w;
            hub += huber_term(xv.x, yv.x) + huber_term(xv.y, yv.y)
                 + huber_term(xv.z, yv.z) + huber_term(xv.w, yv.w);
        }
        x2s[tid] = sx2;
        y2s[tid] = sy2;
#pragma unroll
        for (int m = 16; m >= 1; m >>= 1) hub += __shfl_xor(hub, m, 32);
        if (lane == 0) hred[wave] = hub;
    }

    // ---- init DP boundaries: R[0][*] = R[*][0] = INF, R[0][0] = 0 ---------
    for (int idx = tid; idx < kNT * 17; idx += kThreads) {
        float v = (idx == 0) ? 0.0f : kInf;
        Htop[idx]  = v;
        Vleft[idx] = v;
    }
    __syncthreads();

    if (tid == 0) {
        float s = 0.f;
        for (int i = 0; i < kWaves; ++i) s += hred[i];
        ws[b] = s;                                  // huber partial (sum)
    }

    float* myD = Dts + wave * (16 * 18);
    float* myR = Rt  + wave * (17 * 18);
    const int half = lane >> 4;        // 0: lanes 0-15, 1: lanes 16-31
    const int m16  = lane & 15;

    // ---- tiled wavefront over 63 tile-diagonals ---------------------------
    for (int T = 0; T < 2 * kNT - 1; ++T) {
        const int lo  = (T - (kNT - 1) > 0) ? T - (kNT - 1) : 0;
        const int hi  = (T < kNT - 1) ? T : kNT - 1;
        const int cnt = hi - lo + 1;

#pragma unroll 1
        for (int rep = 0; rep < 2; ++rep) {
            const int idx = wave + rep * kWaves;
            if (idx < cnt) {
                const int ti = lo + idx, tj = T - ti;
                const int gi = ti << 4, gj = tj << 4;

                // Gram tile G = X[gi:gi+16] * Y[gj:gj+16]^T, exact fp32 via
                // 4 chained V_WMMA_F32_16X16X4_F32 (K = 16 in 4 chunks).
                // A 16x4 layout: lane m16, VGPRv = A[m16][k0 + 2*half + v]
                // B  4x16 layout mirrored; C/D: r = v + 8*half, c = m16.
                v8f acc = {};
#pragma unroll
                for (int k0 = 0; k0 < 16; k0 += 4) {
                    v2f av = *(const v2f*)(Xs + (gi + m16) * kXStride + k0 + 2 * half);
                    v2f bv = *(const v2f*)(Ys + (gj + m16) * kXStride + k0 + 2 * half);
                    acc = __builtin_amdgcn_wmma_f32_16x16x4_f32(
                              false, av, false, bv, (short)0, acc, false, false);
                }

                // D tile -> per-wave LDS (stride 18 => conflict-free diagonals).
                // Each half-wave needs the same 8 consecutive x2 values:
                // fetch them with two b128 loads instead of 8 dependent b32s.
                const float4 xa = *(const float4*)(x2s + gi + 8 * half);
                const float4 xb = *(const float4*)(x2s + gi + 8 * half + 4);
                const float  y2c = y2s[gj + m16];
                {
                    const int rbase = 8 * half;
                    myD[(rbase + 0) * 18 + m16] = xa.x + y2c - 2.0f * acc[0];
                    myD[(rbase + 1) * 18 + m16] = xa.y + y2c - 2.0f * acc[1];
                    myD[(rbase + 2) * 18 + m16] = xa.z + y2c - 2.0f * acc[2];
                    myD[(rbase + 3) * 18 + m16] = xa.w + y2c - 2.0f * acc[3];
                    myD[(rbase + 4) * 18 + m16] = xb.x + y2c - 2.0f * acc[4];
                    myD[(rbase + 5) * 18 + m16] = xb.y + y2c - 2.0f * acc[5];
                    myD[(rbase + 6) * 18 + m16] = xb.z + y2c - 2.0f * acc[6];
                    myD[(rbase + 7) * 18 + m16] = xb.w + y2c - 2.0f * acc[7];
                }

                // pull tile boundaries (Rt[0][c] = top row, Rt[r][0] = left col);
                // keep this lane's left-boundary value in a register.
                float vl = 0.0f;
                if (lane < 17) {
                    float hv = Htop[tj * 17 + lane];
                    float vv = Vleft[ti * 17 + lane];
                    myR[lane]          = hv;
                    myR[lane * 18 + 0] = vv;
                    vl = vv;
                }
                LDS_FENCE();

                // 31-step intra-tile DP, wave-local (same-wave DS in-order).
                // Register carry: a(d) == up(d-1) (same LDS address, c+1 per
                // step), lf(d) == own val(d-1) except on the entering c==0
                // step where lf comes from the left boundary.
                float a_reg = 0.0f, val_reg = 0.0f;
#pragma unroll 1
                for (int d = 0; d < 31; ++d) {
                    const int r = lane, c = d - r;
                    if (r < 16 && c >= 0 && c < 16) {
                        float up = myR[r * 18 + c + 1];
                        float dv = myD[r * 18 + c];
                        float a0, lf;
                        if (c == 0) {
                            a0 = vl;                      // R[r][0]
                            lf = myR[(r + 1) * 18];       // R[r+1][0] (enter once)
                        } else {
                            a0 = a_reg;                   // R[r][c] == up(d-1)
                            lf = val_reg;                 // R[r+1][c] == val(d-1)
                        }
                        float mn = fminf(a0, fminf(up, lf));
                        float e  = __expf((mn - a0) * kInvGamma)
                                 + __expf((mn - up) * kInvGamma)
                                 + __expf((mn - lf) * kInvGamma);
                        float val = dv + mn - kGamma * __logf(e);
                        myR[(r + 1) * 18 + c + 1] = val;
                        a_reg   = up;
                        val_reg = val;
                    }
                    __builtin_amdgcn_wave_barrier();
                    LDS_FENCE();
                }

                // publish new boundaries for (ti+1,tj) and (ti,tj+1)
                if (lane < 17) {
                    float nH = myR[16 * 18 + lane];   // bottom row
                    float nV = myR[lane * 18 + 16];   // right col
                    Htop[tj * 17 + lane]  = nH;
                    Vleft[ti * 17 + lane] = nV;
                }
            }
        }
        __syncthreads();   // next tile-diagonal sees updated Htop/Vleft
    }

    if (tid == 0) ws[kB + b] = Htop[31 * 17 + 16];   // R[512][512]
}

__launch_bounds__(64)
__global__ void finalize_kernel(const float* __restrict__ ws,
                                const float* __restrict__ commit,
                                float* __restrict__ out)
{
    __shared__ float sh[kB], ss[kB];
    const int t = threadIdx.x;
    sh[t] = ws[t];
    ss[t] = ws[kB + t];
    __syncthreads();
    if (t == 0) {
        float hs = 0.f, sd = 0.f;
        for (int i = 0; i < kB; ++i) { hs += sh[i]; sd += ss[i]; }
        const float huber = hs / (float)(kB * kL * kC);
        const float sdtw  = (sd / (float)kB) * kAlpha;
        const float cm    = commit[0];
        out[0] = huber + sdtw + cm;
        out[1] = huber;
        out[2] = sdtw;
        out[3] = cm;
    }
}

extern "C" void kernel_launch(void* const* d_in, const int* in_sizes, int n_in,
                              void* d_out, int out_size, void* d_ws, size_t ws_size,
                              hipStream_t stream) {
    (void)in_sizes; (void)n_in; (void)out_size; (void)ws_size;
    const float* x      = (const float*)d_in[0];
    const float* x_hat  = (const float*)d_in[1];
    const float* commit = (const float*)d_in[2];
    float* ws  = (float*)d_ws;     // [0..63] huber partials, [64..127] sdtw per batch
    float* out = (float*)d_out;

    sdtw_huber_kernel<<<kB, kThreads, SMEM_BYTES, stream>>>(x, x_hat, ws);
    finalize_kernel<<<1, 64, 0, stream>>>(ws, commit, out);
}